// SpikingNet_74139725463697
// MI455X (gfx1250) — compile-verified
//
#include <hip/hip_runtime.h>
#include <stddef.h>
#include <stdint.h>

// ---------------------------------------------------------------------------
// SpikingNet forward on gfx1250 (MI455X), fp32 WMMA path with async-LDS
// double-buffered staging.
//   Shapes (from reference setup): B=32, T=512, D=512, H=1024.
//   Layer l: cur = spikes_{l-1} [BT x K] * W_l^T [K x H] + b_l   (WMMA GEMM)
//            spikes_l = LIF_scan_T(cur)                          (serial in T)
// ---------------------------------------------------------------------------

typedef __attribute__((ext_vector_type(2))) float v2f;
typedef __attribute__((ext_vector_type(8))) float v8f;

#define BM 64       // block tile M
#define BN 128      // block tile N
#define BK 16       // K block staged through LDS
#define PAD 20      // LDS row stride in floats: 80B (16B-aligned b128 chunks),
                    // 20*L mod 64 distinct for L=0..15 -> conflict-free reads

// Issue one 16-byte async global->LDS copy (per-lane addresses), ASYNCcnt path.
__device__ __forceinline__ void async_copy16(uint32_t lds_off, const void* gptr) {
    asm volatile("global_load_async_to_lds_b128 %0, %1, off"
                 :: "v"(lds_off), "v"((uint64_t)(uintptr_t)gptr)
                 : "memory");
}
__device__ __forceinline__ void wait_async0() {
    asm volatile("s_wait_asynccnt 0x0" ::: "memory");
}

// out[M,N] = in[M,K] (row-major) * W[N,K]^T + bias[N]
// M % BM == 0, N % BN == 0, K % BK == 0 (true for all three layers).
__global__ __launch_bounds__(256) void gemm_bias_f32(
    const float* __restrict__ in, const float* __restrict__ W,
    const float* __restrict__ bias, float* __restrict__ out,
    int M, int N, int K)
{
    __shared__ __align__(16) float ldsA[2][BM * PAD];   // [stage][m][k]
    __shared__ __align__(16) float ldsB[2][BN * PAD];   // [stage][n][k]

    const int tid  = threadIdx.x;
    const int wave = tid >> 5;
    const int lane = tid & 31;
    const int l16  = lane & 15;
    const int hi   = lane >> 4;        // 0: K pair {0,1}, 1: K pair {2,3}

    const int mBlock = blockIdx.x * BM;
    const int nBlock = blockIdx.y * BN;

    const int mSub = (wave & 3) * 16;  // wave's M offset within block
    const int nSub = (wave >> 2) * 64; // wave's N offset within block

    // LDS byte offsets of the two stages (flat-pointer low 32 bits == LDS offset)
    const uint32_t aBase[2] = { (uint32_t)(uintptr_t)&ldsA[0][0],
                                (uint32_t)(uintptr_t)&ldsA[1][0] };
    const uint32_t bBase[2] = { (uint32_t)(uintptr_t)&ldsB[0][0],
                                (uint32_t)(uintptr_t)&ldsB[1][0] };

    // Per-thread 16B chunk assignments for staging:
    //   A tile: 64 rows x 16 k = 256 chunks, one per thread.
    const int ar = tid >> 2;               // 0..63
    const int ak = (tid & 3) * 4;          // 0,4,8,12
    //   B tile: 128 rows x 16 k = 512 chunks, two per thread (c = tid, tid+256).
    const int br0 = tid >> 2;              // 0..63
    const int bk0 = (tid & 3) * 4;
    const int br1 = (tid + 256) >> 2;      // 64..127
    const int bk1 = bk0;

    v8f acc[4] = {};                       // 16x64 per wave

    const int NB = K / BK;

    // Prologue: stage 0 into buffer 0.
    {
        const int k0 = 0;
        async_copy16(aBase[0] + (uint32_t)(ar * PAD + ak) * 4,
                     in + (size_t)(mBlock + ar) * K + (k0 + ak));
        async_copy16(bBase[0] + (uint32_t)(br0 * PAD + bk0) * 4,
                     W + (size_t)(nBlock + br0) * K + (k0 + bk0));
        async_copy16(bBase[0] + (uint32_t)(br1 * PAD + bk1) * 4,
                     W + (size_t)(nBlock + br1) * K + (k0 + bk1));
    }

    for (int kb = 0; kb < NB; ++kb) {
        const int cur = kb & 1;
        wait_async0();           // stage kb's async copies complete (this wave)
        __syncthreads();         // visible to all; all waves done reading buf[cur^1]

        if (kb + 1 < NB) {       // prefetch stage kb+1 into the other buffer
            const int nxt = cur ^ 1;
            const int k0 = (kb + 1) * BK;
            async_copy16(aBase[nxt] + (uint32_t)(ar * PAD + ak) * 4,
                         in + (size_t)(mBlock + ar) * K + (k0 + ak));
            async_copy16(bBase[nxt] + (uint32_t)(br0 * PAD + bk0) * 4,
                         W + (size_t)(nBlock + br0) * K + (k0 + bk0));
            async_copy16(bBase[nxt] + (uint32_t)(br1 * PAD + bk1) * 4,
                         W + (size_t)(nBlock + br1) * K + (k0 + bk1));
        }

        const float* As = ldsA[cur];
        const float* Bs = ldsB[cur];
        #pragma unroll
        for (int ks = 0; ks < BK; ks += 4) {
            const int kk = ks + hi * 2;  // lanes 0-15: k {ks,ks+1}; 16-31: {ks+2,ks+3}
            const v2f a = *(const v2f*)&As[(mSub + l16) * PAD + kk];
            #pragma unroll
            for (int i = 0; i < 4; ++i) {
                const v2f b = *(const v2f*)&Bs[(nSub + i * 16 + l16) * PAD + kk];
                acc[i] = __builtin_amdgcn_wmma_f32_16x16x4_f32(
                    /*neg_a=*/false, a, /*neg_b=*/false, b,
                    /*c_mod=*/(short)0, acc[i],
                    /*reuse_a=*/false, /*reuse_b=*/false);
            }
        }
    }

    // C/D layout: VGPR r, lanes 0-15 -> (M=r, N=lane); lanes 16-31 -> (M=r+8, N=lane-16)
    const int rowBase = mBlock + mSub + hi * 8;
    #pragma unroll
    for (int i = 0; i < 4; ++i) {
        const int col = nBlock + nSub + i * 16 + l16;
        const float bv = bias[col];
        #pragma unroll
        for (int r = 0; r < 8; ++r) {
            out[(size_t)(rowBase + r) * N + col] = acc[i][r] + bv;
        }
    }
}

// Serial LIF over T for each (b,h): u = 0.5u + cur; s = (u >= 0.5); u = s?0:u.
// cur/out layout: [B, T, H] flat. One thread per (b,h) element.
__global__ __launch_bounds__(256) void lif_scan(
    const float* __restrict__ cur, float* __restrict__ out,
    int B, int T, int H)
{
    const int e = blockIdx.x * blockDim.x + threadIdx.x;  // 0 .. B*H-1 (exact)
    const int b = e / H;
    const int h = e - b * H;
    const size_t base = (size_t)b * T * H + h;

    float u = 0.0f;
    #pragma unroll 4
    for (int t = 0; t < T; ++t) {
        const size_t idx = base + (size_t)t * H;
        u = 0.5f * u + cur[idx];          // DECAY * u + I_t
        const float s = (u >= 0.5f) ? 1.0f : 0.0f;
        out[idx] = s;
        u = (s != 0.0f) ? 0.0f : u;       // hard (multiplicative) reset
    }
}

extern "C" void kernel_launch(void* const* d_in, const int* in_sizes, int n_in,
                              void* d_out, int out_size, void* d_ws, size_t ws_size,
                              hipStream_t stream) {
    // Reference shapes (setup_inputs): B=32, T=512, D=512, H=1024
    const int B = 32, T = 512, D = 512, H = 1024;
    const int M = B * T;                      // 16384 GEMM rows

    const float* x  = (const float*)d_in[0];  // [B,T,D] == [M,D] row-major
    const float* W0 = (const float*)d_in[1];  // [H,D]
    const float* b0 = (const float*)d_in[2];  // [H]
    const float* W1 = (const float*)d_in[3];  // [H,H]
    const float* b1 = (const float*)d_in[4];  // [H]
    const float* W2 = (const float*)d_in[5];  // [H,H]
    const float* b2 = (const float*)d_in[6];  // [H]
    float* outp = (float*)d_out;              // [B,T,H]

    const size_t elems = (size_t)M * H;       // 16,777,216 floats = 64 MiB
    float* ws0 = (float*)d_ws;                // membrane-current buffer
    float* ws1 = ws0 + elems;                 // spike buffer

    dim3 blk(256);
    dim3 gGemm(M / BM, H / BN);               // (256, 8)
    dim3 gScan((B * H) / 256);                // 128 blocks

    // Layer 0: x -> cur -> spikes0
    gemm_bias_f32<<<gGemm, blk, 0, stream>>>(x,   W0, b0, ws0, M, H, D);
    lif_scan    <<<gScan, blk, 0, stream>>>(ws0, ws1, B, T, H);
    // Layer 1: spikes0 -> cur -> spikes1
    gemm_bias_f32<<<gGemm, blk, 0, stream>>>(ws1, W1, b1, ws0, M, H, H);
    lif_scan    <<<gScan, blk, 0, stream>>>(ws0, ws1, B, T, H);
    // Layer 2: spikes1 -> cur -> out spikes
    gemm_bias_f32<<<gGemm, blk, 0, stream>>>(ws1, W2, b2, ws0, M, H, H);
    lif_scan    <<<gScan, blk, 0, stream>>>(ws0, outp, B, T, H);
}